// PoolingNms_22110491640315
// MI455X (gfx1250) — compile-verified
//
#include <hip/hip_runtime.h>
#include <stdint.h>

// Problem geometry (fixed by the reference): (16,1,1080,1920) f32,
// pool-unpool chain k=3,5,6. lcm=30 divides both 1080 and 1920, so the
// op is exactly tileable into independent 30x30 tiles. Each workgroup
// processes a 30x240 strip (8 tiles) fully fused in LDS.
#define IMG_W 1920
#define IMG_H 1080
#define NBATCH 16
#define SW 240          // strip width  (cols)
#define SH 30           // strip height (rows)
#define STRIP (SW * SH) // 7200 floats = 28.8 KB

typedef __attribute__((ext_vector_type(4))) unsigned int u32x4;
typedef __attribute__((ext_vector_type(8))) int i32x8;
typedef __attribute__((ext_vector_type(4))) int i32x4;

#if defined(__has_builtin)
#if __has_builtin(__builtin_amdgcn_tensor_load_to_lds) &&      \
    __has_builtin(__builtin_amdgcn_tensor_store_from_lds) &&   \
    __has_builtin(__builtin_amdgcn_s_wait_tensorcnt)
#define USE_TDM 1
#endif
#endif
#ifndef USE_TDM
#define USE_TDM 0
#endif

#if USE_TDM
// Build a Tensor DMA Descriptor (D#) for a 2D tile:
//   tile SW x SH elements of 4 bytes, row stride IMG_W elements,
//   LDS destination/source contiguous (SW*SH floats).
// Group0: count=1 | lds_addr | global_addr[56:0] | type=2
// Group1: data_size=4B | tensor_dim0=SW | tensor_dim1=SH |
//         tile_dim0=SW | tile_dim1=SH | tensor_dim0_stride=IMG_W
__device__ __forceinline__ void make_tdm_desc(uint64_t gaddr, uint32_t lds_byte,
                                              u32x4 &g0, i32x8 &g1) {
  g0.x = 1u;                                   // count=1, user descriptor
  g0.y = lds_byte;                             // lds_addr (bytes)
  g0.z = (uint32_t)(gaddr & 0xFFFFFFFFu);      // global_addr[31:0]
  g0.w = (uint32_t)((gaddr >> 32) & 0x01FFFFFFu) | (2u << 30); // [56:32] | type=2

  g1[0] = (int)(2u << 16);                     // data_size=2 (4 bytes); wg_mask=0
  g1[1] = (int)(((uint32_t)SW) << 16);         // tensor_dim0[15:0]  (=240)
  g1[2] = (int)(((uint32_t)SH) << 16);         // tensor_dim0 hi=0 | tensor_dim1[15:0]
  g1[3] = (int)(((uint32_t)SW) << 16);         // tensor_dim1 hi=0 | tile_dim0
  g1[4] = (int)SH;                             // tile_dim1 | tile_dim2=0
  g1[5] = (int)IMG_W;                          // tensor_dim0_stride[31:0]
  g1[6] = 0;                                   // stride hi, dim1_stride lo
  g1[7] = 0;
}
#endif

// One pool-unpool stage on a dense SHxSW strip held in LDS.
// Reproduces jnp reference exactly: per KxK tile, max + FIRST-occurrence
// argmax (strict '>' keeps smallest r*K+c), write max at argmax, 0 elsewhere.
template <int K>
__device__ __forceinline__ void pool_unpool_stage(const float *__restrict__ S,
                                                  float *__restrict__ D,
                                                  int tid) {
  constexpr int WX = SW / K;
  constexpr int WY = SH / K;
  constexpr int NW = WX * WY;
  for (int w = tid; w < NW; w += 256) {
    const int wr = w / WX;
    const int wc = w - wr * WX;
    const int base = wr * K * SW + wc * K;
    float best = S[base];
    int bi = 0;
#pragma unroll
    for (int r = 0; r < K; ++r) {
#pragma unroll
      for (int c = 0; c < K; ++c) {
        const float v = S[base + r * SW + c];
        if (v > best) { best = v; bi = r * K + c; }
      }
    }
#pragma unroll
    for (int r = 0; r < K; ++r) {
#pragma unroll
      for (int c = 0; c < K; ++c) {
        D[base + r * SW + c] = ((r * K + c) == bi) ? best : 0.0f;
      }
    }
  }
}

__global__ __launch_bounds__(256) void PoolingNms_fused_kernel(
    const float *__restrict__ x, float *__restrict__ y) {
  __shared__ float bufA[STRIP];
  __shared__ float bufB[STRIP];

  const int tid = threadIdx.x;
  const size_t base =
      ((size_t)blockIdx.z * IMG_H + (size_t)blockIdx.y * SH) * (size_t)IMG_W +
      (size_t)blockIdx.x * SW;

  // ---- Load 30x240 strip into LDS via Tensor Data Mover ----
#if USE_TDM
  if (tid < 32) { // one wave issues the DMA; EXEC ignored by TDM
    u32x4 g0;
    i32x8 g1;
    make_tdm_desc((uint64_t)(uintptr_t)(x + base),
                  (uint32_t)(uintptr_t)(void *)bufA, g0, g1);
    i32x4 z4 = {0, 0, 0, 0};
    i32x8 z8 = {0, 0, 0, 0, 0, 0, 0, 0};
    __builtin_amdgcn_tensor_load_to_lds(g0, g1, z4, z4, z8, 0);
    __builtin_amdgcn_s_wait_tensorcnt(0);
  }
  __syncthreads();
#else
  for (int i = tid; i < STRIP; i += 256) {
    const int r = i / SW;
    bufA[i] = x[base + (size_t)r * IMG_W + (i - r * SW)];
  }
  __syncthreads();
#endif

  // ---- Fused three pool-unpool stages, ping-pong in LDS ----
  pool_unpool_stage<3>(bufA, bufB, tid);
  __syncthreads();
  pool_unpool_stage<5>(bufB, bufA, tid);
  __syncthreads();
  pool_unpool_stage<6>(bufA, bufB, tid);
  __syncthreads();

  // ---- Store strip back via Tensor Data Mover ----
#if USE_TDM
  if (tid < 32) {
    u32x4 g0;
    i32x8 g1;
    make_tdm_desc((uint64_t)(uintptr_t)(y + base),
                  (uint32_t)(uintptr_t)(void *)bufB, g0, g1);
    i32x4 z4 = {0, 0, 0, 0};
    i32x8 z8 = {0, 0, 0, 0, 0, 0, 0, 0};
    __builtin_amdgcn_tensor_store_from_lds(g0, g1, z4, z4, z8, 0);
    __builtin_amdgcn_s_wait_tensorcnt(0);
  }
#else
  for (int i = tid; i < STRIP; i += 256) {
    const int r = i / SW;
    y[base + (size_t)r * IMG_W + (i - r * SW)] = bufB[i];
  }
#endif
}

extern "C" void kernel_launch(void *const *d_in, const int *in_sizes, int n_in,
                              void *d_out, int out_size, void *d_ws,
                              size_t ws_size, hipStream_t stream) {
  (void)in_sizes; (void)n_in; (void)out_size; (void)d_ws; (void)ws_size;
  const float *x = (const float *)d_in[0];
  float *y = (float *)d_out;
  dim3 grid(IMG_W / SW, IMG_H / SH, NBATCH); // (8, 36, 16)
  PoolingNms_fused_kernel<<<grid, 256, 0, stream>>>(x, y);
}